// Sampler_81046032876033
// MI455X (gfx1250) — compile-verified
//
#include <hip/hip_runtime.h>
#include <hip/hip_bf16.h>
#include <stdint.h>

// B=256 rows, V=128000 vocab. One 512-thread workgroup per row (wave32 -> 16 waves).
// Pass 1: async global->LDS (GLOBAL_LOAD_ASYNC_TO_LDS_B128) double-buffered staging,
//         per-wave private stage regions (no barriers in steady state), computing
//         row max + 4096-bin key histogram. HBM-bound: ~5.5us @ 23.3 TB/s.
// Pass 2: sum-exp + candidate collection (row is L2-resident: 128MB < 192MB L2).
// Then LDS bitonic sort of <=1024 candidates, serial top-100 mask + sample.

#define TPB       512
#define NBINS     4096
#define BINS_PER  (NBINS / TPB)   // 8
#define CAP       1024
#define KTOP      100

__device__ __forceinline__ uint32_t fkey(float f) {
    // order-preserving float -> uint map (ascending)
    uint32_t u = __float_as_uint(f);
    return (u & 0x80000000u) ? ~u : (u | 0x80000000u);
}
__device__ __forceinline__ float fkey_inv(uint32_t k) {
    uint32_t u = (k & 0x80000000u) ? (k & 0x7FFFFFFFu) : ~k;
    return __uint_as_float(u);
}

// Async DMA: 16 bytes/lane, global -> LDS, tracked by ASYNCcnt (not LOADcnt).
__device__ __forceinline__ void async_ld16(uint32_t lds_off, unsigned long long gaddr) {
    asm volatile("global_load_async_to_lds_b128 %0, %1, off"
                 :: "v"(lds_off), "v"(gaddr) : "memory");
}

__global__ __launch_bounds__(TPB) void Sampler_81046032876033_kernel(
    const float* __restrict__ logits,
    const float* __restrict__ temperatures,
    const float* __restrict__ top_ps,
    const int*   __restrict__ top_ks,
    int*         __restrict__ out,
    int V)
{
    __shared__ __align__(16) float stage[2][TPB * 4];   // 16 KB double-buffer stage
    __shared__ uint32_t hist[NBINS];
    __shared__ uint32_t part[TPB];
    __shared__ uint32_t part2[TPB];
    __shared__ unsigned long long comp[CAP];
    __shared__ float    red[TPB];
    __shared__ float    w[KTOP];
    __shared__ uint32_t s_thresh;
    __shared__ uint32_t s_cnt;
    __shared__ int      s_coarse;
    __shared__ float    s_M, s_S;

    const int row = blockIdx.x;
    const int tid = threadIdx.x;
    const float* __restrict__ rowp = logits + (size_t)row * (size_t)V;
    const float4* __restrict__ rp4 = (const float4*)rowp;
    const int V4 = V >> 2;

    for (int i = tid; i < NBINS; i += TPB) hist[i] = 0u;
    if (tid == 0) s_cnt = 0u;
    __syncthreads();

    // ---------------- Pass 1: async-staged row max + 12-bit key histogram -------
    // Each lane owns one float4 slot per buffer; each wave waits only on its own
    // async loads and reads only LDS it wrote -> no block barrier in the loop.
    const uint32_t lds0 = (uint32_t)(uintptr_t)&stage[0][tid * 4];
    const uint32_t lds1 = (uint32_t)(uintptr_t)&stage[1][tid * 4];

    float lmax = -INFINITY;
    const int NIT = V4 / TPB;                 // 62 full block-iterations
    if (NIT > 0)
        async_ld16(lds0, (unsigned long long)(uintptr_t)&rp4[tid]);
    for (int it = 0; it < NIT; ++it) {
        if (it + 1 < NIT) {
            async_ld16((it + 1) & 1 ? lds1 : lds0,
                       (unsigned long long)(uintptr_t)&rp4[(it + 1) * TPB + tid]);
            asm volatile("s_wait_asynccnt 0x1" ::: "memory");  // current chunk done
        } else {
            asm volatile("s_wait_asynccnt 0x0" ::: "memory");  // drain
        }
        const float4 x = *(const float4*)((it & 1) ? &stage[1][tid * 4]
                                                   : &stage[0][tid * 4]);
        lmax = fmaxf(lmax, fmaxf(fmaxf(x.x, x.y), fmaxf(x.z, x.w)));
        atomicAdd(&hist[fkey(x.x) >> 20], 1u);
        atomicAdd(&hist[fkey(x.y) >> 20], 1u);
        atomicAdd(&hist[fkey(x.z) >> 20], 1u);
        atomicAdd(&hist[fkey(x.w) >> 20], 1u);
    }
    for (int i = NIT * TPB + tid; i < V4; i += TPB) {   // float4 remainder
        float4 x = rp4[i];
        lmax = fmaxf(lmax, fmaxf(fmaxf(x.x, x.y), fmaxf(x.z, x.w)));
        atomicAdd(&hist[fkey(x.x) >> 20], 1u);
        atomicAdd(&hist[fkey(x.y) >> 20], 1u);
        atomicAdd(&hist[fkey(x.z) >> 20], 1u);
        atomicAdd(&hist[fkey(x.w) >> 20], 1u);
    }
    for (int i = (V4 << 2) + tid; i < V; i += TPB) {    // scalar tail (empty here)
        float x = rowp[i];
        lmax = fmaxf(lmax, x);
        atomicAdd(&hist[fkey(x) >> 20], 1u);
    }

    // deterministic max reduction
    red[tid] = lmax;
    __syncthreads();
    for (int s = TPB >> 1; s > 0; s >>= 1) {
        if (tid < s) red[tid] = fmaxf(red[tid], red[tid + s]);
        __syncthreads();
    }
    if (tid == 0) s_M = red[0];
    __syncthreads();

    // ---------------- Threshold: largest bin t with count(key12 >= t) >= KTOP --
    {
        uint32_t acc = 0;
        const int base = tid * BINS_PER;
        #pragma unroll
        for (int j = 0; j < BINS_PER; ++j) acc += hist[base + j];
        part[tid] = acc;
    }
    __syncthreads();
    // inclusive suffix scan over the 512 coarse partials (Hillis-Steele, ping-pong)
    {
        uint32_t* src = part;
        uint32_t* dst = part2;
        for (int off = 1; off < TPB; off <<= 1) {
            uint32_t v = src[tid];
            if (tid + off < TPB) v += src[tid + off];
            dst[tid] = v;
            __syncthreads();
            uint32_t* t = src; src = dst; dst = t;
        }
        // src[t] = #elements with coarse bin >= t (non-increasing in t)
        if (src[tid] >= KTOP && (tid == TPB - 1 || src[tid + 1] < KTOP)) s_coarse = tid;
        __syncthreads();
        if (tid == 0) {
            int c = s_coarse;
            uint32_t above = (c == TPB - 1) ? 0u : src[c + 1];
            uint32_t acc = above;
            int t = c * BINS_PER;
            for (int b = c * BINS_PER + BINS_PER - 1; b >= c * BINS_PER; --b) {
                acc += hist[b];
                if (acc >= KTOP) { t = b; break; }
            }
            s_thresh = (uint32_t)t;
        }
    }
    __syncthreads();

    // ---------------- Pass 2: sum of exp((x-M)/T) + collect candidates ----------
    const float M    = s_M;
    const float invT = 1.0f / temperatures[row];
    const uint32_t thr = s_thresh;
    float lsum = 0.0f;
    for (int i = tid; i < V4; i += TPB) {
        float4 x = rp4[i];
        lsum += __expf((x.x - M) * invT);
        lsum += __expf((x.y - M) * invT);
        lsum += __expf((x.z - M) * invT);
        lsum += __expf((x.w - M) * invT);
        const int ib = i << 2;
        #pragma unroll
        for (int c = 0; c < 4; ++c) {
            float v = (c == 0) ? x.x : (c == 1) ? x.y : (c == 2) ? x.z : x.w;
            uint32_t k = fkey(v);
            if ((k >> 20) >= thr) {
                uint32_t p = atomicAdd(&s_cnt, 1u);
                if (p < CAP)
                    comp[p] = ((unsigned long long)k << 32) | (uint32_t)(~(uint32_t)(ib + c));
            }
        }
    }
    for (int i = (V4 << 2) + tid; i < V; i += TPB) {
        float v = rowp[i];
        lsum += __expf((v - M) * invT);
        uint32_t k = fkey(v);
        if ((k >> 20) >= thr) {
            uint32_t p = atomicAdd(&s_cnt, 1u);
            if (p < CAP)
                comp[p] = ((unsigned long long)k << 32) | (uint32_t)(~(uint32_t)i);
        }
    }

    // deterministic sum reduction
    red[tid] = lsum;
    __syncthreads();
    for (int s = TPB >> 1; s > 0; s >>= 1) {
        if (tid < s) red[tid] += red[tid + s];
        __syncthreads();
    }
    if (tid == 0) s_S = red[0];
    __syncthreads();

    // pad unused candidate slots (sort to the bottom)
    const uint32_t ncand = (s_cnt < CAP) ? s_cnt : CAP;
    for (int i = tid; i < CAP; i += TPB)
        if ((uint32_t)i >= ncand) comp[i] = 0ull;
    __syncthreads();

    // ---------------- Bitonic sort, descending (value desc, index asc) ----------
    for (int k = 2; k <= CAP; k <<= 1) {
        for (int j = k >> 1; j > 0; j >>= 1) {
            for (int i = tid; i < CAP; i += TPB) {
                int ixj = i ^ j;
                if (ixj > i) {
                    unsigned long long a = comp[i], b = comp[ixj];
                    bool doswap = ((i & k) == 0) ? (a < b) : (a > b);
                    if (doswap) { comp[i] = b; comp[ixj] = a; }
                }
            }
            __syncthreads();
        }
    }

    // ---------------- Finalize: mask top-k / top-p over top-100, sample ---------
    if (tid == 0) {
        const int   K2 = (int)((ncand < (uint32_t)KTOP) ? ncand : (uint32_t)KTOP);
        const float S  = s_S;
        const int   tk = top_ks[row];
        const float tp = top_ps[row];

        float csum = 0.0f, wsum = 0.0f;
        for (int i = 0; i < K2; ++i) {
            uint32_t key = (uint32_t)(comp[i] >> 32);
            float p = __expf((fkey_inv(key) - M) * invT) / S;
            csum += p;                                     // cumsum BEFORE masking
            float wi = ((i < tk) && (csum - p <= tp)) ? p : 0.0f;
            w[i] = wi;
            wsum += wi;
        }

        // deterministic per-row uniform (stand-in for the threefry stream)
        uint32_t h = (uint32_t)row * 0x9E3779B9u + 42u;
        h ^= h >> 16; h *= 0x7FEB352Du;
        h ^= h >> 15; h *= 0x846CA68Bu;
        h ^= h >> 16;
        float u = (float)(h >> 8) * (1.0f / 16777216.0f);
        float target = u * wsum;

        int pick = -1, lastnz = 0;
        float acc = 0.0f;
        for (int i = 0; i < K2; ++i) {
            if (w[i] > 0.0f) lastnz = i;
            acc += w[i];
            if (pick < 0 && acc > target) pick = i;
        }
        if (pick < 0) pick = lastnz;

        out[row] = (int)(~(uint32_t)(comp[pick] & 0xFFFFFFFFu));
    }
}

extern "C" void kernel_launch(void* const* d_in, const int* in_sizes, int n_in,
                              void* d_out, int out_size, void* d_ws, size_t ws_size,
                              hipStream_t stream) {
    const float* logits = (const float*)d_in[0];
    const float* temps  = (const float*)d_in[1];
    const float* tps    = (const float*)d_in[2];
    const int*   tks    = (const int*)d_in[3];
    // d_in[4] (min_ps) unused: need_min_p_sampling=False in the reference path.
    const int B = in_sizes[1];               // 256
    const int V = in_sizes[0] / B;           // 128000
    Sampler_81046032876033_kernel<<<B, TPB, 0, stream>>>(
        logits, temps, tps, tks, (int*)d_out, V);
}